// SphericalHarmonics_17231408792195
// MI455X (gfx1250) — compile-verified
//
#include <hip/hip_runtime.h>

// ---------------------------------------------------------------------------
// Real spherical harmonics, L=10 (100 outputs/point), N=1e6 points.
// Roofline: 400MB out @ 23.3TB/s => ~17.5us floor; compute ~1 GFLOP (free).
// Legendre polys = constant 55x10 coeff matrix (monomial basis, normalization
// folded at compile time) x per-point x-powers via V_WMMA_F32_16X16X4_F32.
// B-matrix powers are built with a branchless multiply chain (no arrays, no
// select trees). Readout branches once on wave-half so every index is
// constexpr. Output staged in LDS, written as coalesced float4 stores.
// ---------------------------------------------------------------------------

typedef __attribute__((ext_vector_type(2))) float v2f;
typedef __attribute__((ext_vector_type(8))) float v8f;

// ----- compile-time coefficient generation (double precision) --------------
constexpr double CPI = 3.141592653589793238462643383279502884;

constexpr double csqrt(double x) {
  if (x <= 0.0) return 0.0;
  double g = x > 1.0 ? x : 1.0;
  for (int i = 0; i < 100; ++i) g = 0.5 * (g + x / g);
  return g;
}
constexpr double cfact(int n) {
  double r = 1.0;
  for (int i = 2; i <= n; ++i) r *= (double)i;
  return r;
}

struct CoefPack {
  float coef[64][12];  // rows r=0..54 are (l,m) pairs l-major; k padded to 12
  int   m[64];
  int   jp[64];        // out column for +m : l*l + l + m
  int   jm[64];        // out column for -m : l*l + l - m
};

constexpr CoefPack build() {
  CoefPack P{};
  // poly[l][m][k]: coefficients of p_hat_{l,m}(x) = P_l^m(x) / s^m  (CS phase)
  double poly[10][10][10] = {};
  double dd = 1.0;
  for (int m = 0; m < 10; ++m) {
    if (m > 0) dd *= -(2.0 * m - 1.0);
    poly[m][m][0] = dd;
    if (m + 1 < 10) {
      for (int k = 0; k + 1 < 10; ++k)
        poly[m + 1][m][k + 1] = (2.0 * m + 1.0) * poly[m][m][k];
    }
    for (int l = m + 2; l < 10; ++l) {
      for (int k = 0; k < 10; ++k) {
        double t = 0.0;
        if (k >= 1) t += (2.0 * l - 1.0) * poly[l - 1][m][k - 1];
        t -= (l + m - 1.0) * poly[l - 2][m][k];
        poly[l][m][k] = t / (double)(l - m);
      }
    }
  }
  int r = 0;
  for (int l = 0; l < 10; ++l) {
    for (int mm = 0; mm <= l; ++mm) {
      double K = csqrt((2.0 * l + 1.0) / (4.0 * CPI) * cfact(l - mm) / cfact(l + mm));
      double mult = (mm == 0) ? K : csqrt(2.0) * K;
      for (int k = 0; k < 12; ++k)
        P.coef[r][k] = (k < 10) ? (float)(mult * poly[l][mm][k]) : 0.0f;
      P.m[r]  = mm;
      P.jp[r] = l * l + l + mm;
      P.jm[r] = l * l + l - mm;
      ++r;
    }
  }
  for (; r < 64; ++r) {
    P.m[r] = 0; P.jp[r] = 0; P.jm[r] = 0;
    for (int k = 0; k < 12; ++k) P.coef[r][k] = 0.0f;
  }
  return P;
}

constexpr CoefPack CPC = build();     // compile-time metadata (constant-folded)
__constant__ CoefPack g_cp = build(); // runtime A-matrix coefficients

// ----- static unroll helper -------------------------------------------------
template <int N, int I = 0, class F>
__device__ inline void static_for(F f) {
  if constexpr (I < N) {
    f.template operator()<I>();
    static_for<N, I + 1, F>(f);
  }
}

// Readout of one wave-half: every index is a compile-time constant, so LDS
// stores become immediate-offset ds_store from one base pointer.
template <int HI>
__device__ inline void readout_half(const v8f (&acc)[4], float* __restrict__ row,
                                    const float (&Ac)[10], const float (&As)[10]) {
  static_for<4>([&]<int T>() {
    static_for<8>([&]<int V>() {
      constexpr int r = T * 16 + HI * 8 + V;
      if constexpr (r < 55) {
        constexpr int M  = CPC.m[r];
        constexpr int JP = CPC.jp[r];
        constexpr int JM = CPC.jm[r];
        float ph = acc[T][V];
        row[JP] = ph * Ac[M];
        if constexpr (M > 0) row[JM] = ph * As[M];
      }
    });
  });
}

// ---------------------------------------------------------------------------
__global__ __launch_bounds__(128) void sh_wmma_kernel(
    const float2* __restrict__ lonlat, float* __restrict__ out, long long Np) {
  __shared__ float lds[64 * 101];

  const int tid  = (int)threadIdx.x & 127;   // range hint for the optimizer
  const int wave = tid >> 5;
  const int lane = tid & 31;
  const int hi   = lane >> 4;    // wave half (K/M split of the WMMA layout)
  const int n    = lane & 15;    // point index within this wave's 16-pt group
  const int pl   = wave * 16 + n;                       // local point 0..63
  long long p    = (long long)blockIdx.x * 64 + pl;     // global point
  long long pc   = p < Np ? p : (Np - 1);               // clamp (keep EXEC full)

  float2 ll = lonlat[pc];
  const float D2R = 0.017453292519943295f;
  float phi = (ll.x + 180.0f) * D2R;   // [0, 2pi]
  float th  = (ll.y + 90.0f)  * D2R;   // [0, pi]
  float s, x;   __sincosf(th,  &s,  &x);   // s = sin(theta) >= 0, x = cos(theta)
  float s1, c1; __sincosf(phi, &s1, &c1);

  // Per-lane B-matrix powers: lane needs x^(4*KT + 2*hi) and the next power.
  // One cndmask, then a pure multiply chain. High half at KT=2 produces
  // x^10/x^11, which multiply compile-time-zero A columns -> contribute 0.
  float x2 = x * x;
  float x4 = x2 * x2;
  float e[6];
  e[0] = hi ? x2 : 1.0f;
  e[1] = e[0] * x;
  e[2] = e[0] * x4;
  e[3] = e[1] * x4;
  e[4] = e[2] * x4;
  e[5] = e[3] * x4;

  // Ac[m] = s^m cos(m phi), As[m] = s^m sin(m phi) (prefolded trig factors)
  float Ac[10], As[10];
  {
    float spm = 1.0f, cmv = 1.0f, smv = 0.0f;
    Ac[0] = 1.0f; As[0] = 0.0f;
#pragma unroll
    for (int m = 1; m < 10; ++m) {
      float cn = cmv * c1 - smv * s1;
      float sn = smv * c1 + cmv * s1;
      cmv = cn; smv = sn;
      spm *= s;
      Ac[m] = spm * cn;
      As[m] = spm * sn;
    }
  }

  // 12 back-to-back WMMAs: 4 M-tiles (16 coeff rows) x 3 K-tiles (4 monomials)
  const int koff = hi ? 2 : 0;
  v8f acc[4];
  static_for<4>([&]<int T>() {
    acc[T] = (v8f){0.f, 0.f, 0.f, 0.f, 0.f, 0.f, 0.f, 0.f};
    const v2f* crow2 = (const v2f*)(&g_cp.coef[T * 16 + n][koff]);  // 8B aligned
    static_for<3>([&]<int KT>() {
      v2f a = crow2[KT * 2];
      v2f b;
      b.x = e[2 * KT];
      b.y = e[2 * KT + 1];
      acc[T] = __builtin_amdgcn_wmma_f32_16x16x4_f32(
          false, a, false, b, (short)0, acc[T], false, false);
    });
  });

  // Single divergent branch; each body is fully constexpr-indexed.
  float* row = &lds[pl * 101];
  if (hi) readout_half<1>(acc, row, Ac, As);
  else    readout_half<0>(acc, row, Ac, As);

  __syncthreads();

  // Coalesced copy-out: 64 points * 100 f32 = 6400 f32 contiguous per block.
  long long blk_base = (long long)blockIdx.x * 64;
  float* dst = out + blk_base * 100;
  int valid_pts = (int)((Np - blk_base) < 64 ? (Np - blk_base) : 64);
  if (valid_pts == 64) {
    // Full block (the only case when N % 64 == 0): 12 unguarded chunks + tail.
#pragma unroll
    for (int c = 0; c < 12; ++c) {
      int flat = (c * 128 + tid) * 4;       // < 6400 for c <= 11, tid <= 127
      int pq = flat / 100;
      int j  = flat - pq * 100;
      const float* src = &lds[pq * 101 + j];
      float4 vv = {src[0], src[1], src[2], src[3]};
      *(float4*)(dst + flat) = vv;
    }
    {
      int flat = (12 * 128 + tid) * 4;
      if (flat < 6400) {
        int pq = flat / 100;
        int j  = flat - pq * 100;
        const float* src = &lds[pq * 101 + j];
        float4 vv = {src[0], src[1], src[2], src[3]};
        *(float4*)(dst + flat) = vv;
      }
    }
  } else {
    int total = valid_pts * 100;
#pragma unroll
    for (int c = 0; c < 13; ++c) {
      int flat = (c * 128 + tid) * 4;
      if (flat < total) {
        int pq = flat / 100;
        int j  = flat - pq * 100;
        const float* src = &lds[pq * 101 + j];
        float4 vv = {src[0], src[1], src[2], src[3]};
        *(float4*)(dst + flat) = vv;
      }
    }
  }
}

// ---------------------------------------------------------------------------
extern "C" void kernel_launch(void* const* d_in, const int* in_sizes, int n_in,
                              void* d_out, int out_size, void* d_ws, size_t ws_size,
                              hipStream_t stream) {
  (void)n_in; (void)out_size; (void)d_ws; (void)ws_size;
  const float2* lonlat = (const float2*)d_in[0];
  float* out = (float*)d_out;
  long long n = (long long)in_sizes[0] / 2;   // [N,2] flat
  int blocks = (int)((n + 63) / 64);          // 64 points per 128-thread block
  sh_wmma_kernel<<<blocks, 128, 0, stream>>>(lonlat, out, n);
}